// RadiusConsistencyLoss_28484223107627
// MI455X (gfx1250) — compile-verified
//
#include <hip/hip_runtime.h>

typedef float v2f __attribute__((ext_vector_type(2)));
typedef float v4f __attribute__((ext_vector_type(4)));
typedef float v8f __attribute__((ext_vector_type(8)));

#define BB     2
#define NPTS   32768            // N
#define NCENT  1024             // NC
#define SPC    32               // S
#define NSP    (NCENT * SPC)    // sphere points per batch = 32768
#define NGRP   (NPTS / 16)      // 2048 row/col groups of 16
#define CGRP   (NCENT / 16)     // 64 center groups
#define BPAD   256              // floats: 2 tile-pairs of pipeline over-read pad

// ---------------------------------------------------------------------------
// A-pack: WMMA 16x4 A rows, one float2 per lane per 16-group.
//   lane<16:(-2x,-2y)   lane>=16:(-2z, 1)      norms[p] = |p|^2
// ---------------------------------------------------------------------------
__global__ void pack_A(const float* __restrict__ src,
                       float* __restrict__ packed,
                       float* __restrict__ norms, int total) {
  int p = blockIdx.x * blockDim.x + threadIdx.x;
  if (p >= total) return;
  float x = src[3 * p + 0], y = src[3 * p + 1], z = src[3 * p + 2];
  norms[p] = x * x + y * y + z * z;
  int g = p >> 4, r = p & 15;
  float2* pp = (float2*)packed;
  pp[g * 32 + r]      = make_float2(-2.f * x, -2.f * y);
  pp[g * 32 + 16 + r] = make_float2(-2.f * z, 1.f);
}

// ---------------------------------------------------------------------------
// B-pack, tile-PAIR interleaved: lane l of pair-group g2 holds float4
// [(g2*32+l)]: .xy = even tile pair, .zw = odd tile pair.
//   lane<16:( x,  y)   lane>=16:( z, |p|^2)
// One global_load_b128 per lane feeds two WMMAs.
// ---------------------------------------------------------------------------
__global__ void pack_B(const float* __restrict__ src,
                       float* __restrict__ packed, int total) {
  int p = blockIdx.x * blockDim.x + threadIdx.x;
  if (p >= total) return;
  float x = src[3 * p + 0], y = src[3 * p + 1], z = src[3 * p + 2];
  float n2 = x * x + y * y + z * z;
  int g = p >> 4, r = p & 15;
  int g2 = g >> 1, q = g & 1;
  float2* pp = (float2*)packed;
  pp[((g2 * 32 + r) << 1) + q]      = make_float2(x, y);
  pp[((g2 * 32 + 16 + r) << 1) + q] = make_float2(z, n2);
}

__global__ void init_ws(int* __restrict__ maxd, float* __restrict__ p1sum) {
  int i = blockIdx.x * blockDim.x + threadIdx.x;
  if (i < BB * NCENT) maxd[i] = 0;
  if (i < BB) p1sum[i] = 0.f;
}

// ---------------------------------------------------------------------------
// Part 1: per xyz point, min/argmin squared distance over 1024 centers.
// WMMA with C = |p|^2 accumulator yields d^2 directly.
// ---------------------------------------------------------------------------
__global__ void __launch_bounds__(256)
part1_kernel(const float* __restrict__ xyzA, const float* __restrict__ xyzn,
             const float* __restrict__ cB, const float* __restrict__ radius,
             float* __restrict__ p1sum) {
  const int lane = threadIdx.x & 31;
  const int w = blockIdx.x * (blockDim.x >> 5) + (threadIdx.x >> 5);
  const int b = w >> 11;
  const int g = w & (NGRP - 1);
  const int lh = lane >> 4;
  const int ln = lane & 15;

  const v2f* ap = (const v2f*)xyzA;
  v2f a = ap[(b * NGRP + g) * 32 + lane];

  v8f cacc;                       // C[M][N] = |p_M|^2  (M = v + 8*lh)
#pragma unroll
  for (int v = 0; v < 8; ++v)
    cacc[v] = xyzn[b * NPTS + g * 16 + v + 8 * lh];

  float rmin[8];
  int ridx[8];
#pragma unroll
  for (int v = 0; v < 8; ++v) { rmin[v] = 3.0e38f; ridx[v] = 0; }

  const int N2 = CGRP / 2;        // 32 tile pairs
  const v4f* bp4 = (const v4f*)cB + (size_t)b * N2 * 32 + lane;

  v4f cur = bp4[0];
  v4f nxt = bp4[32];
#pragma unroll 2
  for (int t2 = 0; t2 < N2; ++t2) {
    v4f fut = bp4[(t2 + 2) * 32];         // unconditional: buffer padded
    v2f b0 = __builtin_shufflevector(cur, cur, 0, 1);
    v2f b1 = __builtin_shufflevector(cur, cur, 2, 3);
    v8f d0 = __builtin_amdgcn_wmma_f32_16x16x4_f32(
        false, a, false, b0, (short)0, cacc, false, false);
    v8f d1 = __builtin_amdgcn_wmma_f32_16x16x4_f32(
        false, a, false, b1, (short)0, cacc, false, false);
    int i0 = t2 * 32 + ln, i1 = t2 * 32 + 16 + ln;
#pragma unroll
    for (int v = 0; v < 8; ++v) {
      if (d0[v] < rmin[v]) { rmin[v] = d0[v]; ridx[v] = i0; }
      if (d1[v] < rmin[v]) { rmin[v] = d1[v]; ridx[v] = i1; }
    }
    cur = nxt; nxt = fut;
  }

  // min+argmin across the 16 column lanes of each half (wave32 shuffles)
#pragma unroll
  for (int off = 1; off < 16; off <<= 1) {
#pragma unroll
    for (int v = 0; v < 8; ++v) {
      float ov = __shfl_xor(rmin[v], off, 32);
      int   oi = __shfl_xor(ridx[v], off, 32);
      if (ov < rmin[v]) { rmin[v] = ov; ridx[v] = oi; }
    }
  }

  if (ln == 0) {  // lanes 0 (M=0..7) and 16 (M=8..15)
    float local = 0.f;
#pragma unroll
    for (int v = 0; v < 8; ++v) {
      float d = sqrtf(fmaxf(rmin[v], 1e-12f));
      float r = radius[b * NCENT + ridx[v]];
      local += fmaxf(d - r, 0.f);   // |min(r-d, 0)|
    }
    atomicAdd(&p1sum[b], local);
  }
}

// ---------------------------------------------------------------------------
// Part 2: per sphere point, min squared distance over all 32768 xyz points;
// per-center max over its 32 samples via one atomicMax per wave.
// ---------------------------------------------------------------------------
__global__ void __launch_bounds__(256)
part2_kernel(const float* __restrict__ spA, const float* __restrict__ spn,
             const float* __restrict__ xyzB, int* __restrict__ maxd) {
  const int lane = threadIdx.x & 31;
  const int w = blockIdx.x * (blockDim.x >> 5) + (threadIdx.x >> 5);
  const int b = w >> 11;
  const int g = w & (NGRP - 1);
  const int lh = lane >> 4;

  const v2f* ap = (const v2f*)spA;
  v2f a = ap[(b * NGRP + g) * 32 + lane];

  v8f cacc;                       // C[M][N] = |sp_M|^2
#pragma unroll
  for (int v = 0; v < 8; ++v)
    cacc[v] = spn[b * NSP + g * 16 + v + 8 * lh];

  float rmin[8];
#pragma unroll
  for (int v = 0; v < 8; ++v) rmin[v] = 3.0e38f;

  const int N2 = NGRP / 2;        // 1024 tile pairs
  const v4f* bp4 = (const v4f*)xyzB + (size_t)b * N2 * 32 + lane;

  v4f cur = bp4[0];
  v4f nxt = bp4[32];
#pragma unroll 2
  for (int t2 = 0; t2 < N2; ++t2) {
    v4f fut = bp4[(t2 + 2) * 32];         // unconditional: buffer padded
    v2f b0 = __builtin_shufflevector(cur, cur, 0, 1);
    v2f b1 = __builtin_shufflevector(cur, cur, 2, 3);
    v8f d0 = __builtin_amdgcn_wmma_f32_16x16x4_f32(
        false, a, false, b0, (short)0, cacc, false, false);
    v8f d1 = __builtin_amdgcn_wmma_f32_16x16x4_f32(
        false, a, false, b1, (short)0, cacc, false, false);
#pragma unroll
    for (int v = 0; v < 8; ++v)           // v_min3_num_f32 pattern
      rmin[v] = fminf(fminf(rmin[v], d0[v]), d1[v]);
    cur = nxt; nxt = fut;
  }

#pragma unroll
  for (int off = 1; off < 16; off <<= 1) {
#pragma unroll
    for (int v = 0; v < 8; ++v)
      rmin[v] = fminf(rmin[v], __shfl_xor(rmin[v], off, 32));
  }

  float dmax = 0.f;
#pragma unroll
  for (int v = 0; v < 8; ++v)
    dmax = fmaxf(dmax, sqrtf(fmaxf(rmin[v], 1e-12f)));
  dmax = fmaxf(dmax, __shfl_xor(dmax, 16, 32));  // combine both halves

  if (lane == 0)   // positive floats: int compare is monotonic, init = 0
    atomicMax(&maxd[b * NCENT + (g >> 1)], __float_as_int(dmax));
}

// ---------------------------------------------------------------------------
// Final scalar: mean_b( mean_c(max_d) + p1sum/N )
// ---------------------------------------------------------------------------
__global__ void __launch_bounds__(256)
final_kernel(const int* __restrict__ maxd, const float* __restrict__ p1sum,
             float* __restrict__ out) {
  __shared__ float s0[256], s1[256];
  int tid = threadIdx.x;
  float a0 = 0.f, a1 = 0.f;
  for (int c = tid; c < NCENT; c += 256) {
    a0 += __int_as_float(maxd[c]);
    a1 += __int_as_float(maxd[NCENT + c]);
  }
  s0[tid] = a0; s1[tid] = a1;
  __syncthreads();
  for (int s = 128; s > 0; s >>= 1) {
    if (tid < s) { s0[tid] += s0[tid + s]; s1[tid] += s1[tid + s]; }
    __syncthreads();
  }
  if (tid == 0) {
    float r0 = s0[0] / (float)NCENT + p1sum[0] / (float)NPTS;
    float r1 = s1[0] / (float)NCENT + p1sum[1] / (float)NPTS;
    out[0] = 0.5f * (r0 + r1);
  }
}

// ---------------------------------------------------------------------------
extern "C" void kernel_launch(void* const* d_in, const int* in_sizes, int n_in,
                              void* d_out, int out_size, void* d_ws, size_t ws_size,
                              hipStream_t stream) {
  const float* centers = (const float*)d_in[0];   // [B,NC,3]
  const float* radius  = (const float*)d_in[1];   // [B,NC,1]
  const float* xyz     = (const float*)d_in[2];   // [B,N,3]
  const float* sph     = (const float*)d_in[3];   // [B,NC*S,3]

  float* ws    = (float*)d_ws;
  float* xyzA  = ws;                               // B*N*4
  float* spA   = xyzA + (size_t)BB * NPTS * 4;     // B*NSP*4
  float* xyzB  = spA  + (size_t)BB * NSP * 4;      // B*N*4 (+pad)
  float* cB    = xyzB + (size_t)BB * NPTS * 4 + BPAD;   // B*NC*4 (+pad)
  float* xyzn  = cB   + (size_t)BB * NCENT * 4 + BPAD;  // B*N
  float* spn   = xyzn + (size_t)BB * NPTS;         // B*NSP
  int*   maxd  = (int*)(spn + (size_t)BB * NSP);   // B*NC
  float* p1sum = (float*)(maxd + BB * NCENT);      // B

  pack_A<<<(BB * NPTS + 255) / 256, 256, 0, stream>>>(xyz, xyzA, xyzn, BB * NPTS);
  pack_A<<<(BB * NSP + 255) / 256, 256, 0, stream>>>(sph, spA, spn, BB * NSP);
  pack_B<<<(BB * NPTS + 255) / 256, 256, 0, stream>>>(xyz, xyzB, BB * NPTS);
  pack_B<<<(BB * NCENT + 255) / 256, 256, 0, stream>>>(centers, cB, BB * NCENT);
  init_ws<<<(BB * NCENT + 255) / 256, 256, 0, stream>>>(maxd, p1sum);

  // 4096 waves each: 512 blocks x 8 waves (wave32, 256 threads)
  part1_kernel<<<BB * NGRP / 8, 256, 0, stream>>>(xyzA, xyzn, cB, radius, p1sum);
  part2_kernel<<<BB * NGRP / 8, 256, 0, stream>>>(spA, spn, xyzB, maxd);
  final_kernel<<<1, 256, 0, stream>>>(maxd, p1sum, (float*)d_out);
}